// CGLayer_72138270703996
// MI455X (gfx1250) — compile-verified
//
#include <hip/hip_runtime.h>
#include <hip/hip_bf16.h>
#include <math.h>

typedef float v2f __attribute__((ext_vector_type(2)));
typedef float v8f __attribute__((ext_vector_type(8)));

#define B_DIM 8
#define N_DIM 128
#define C_DIM 16
#define MTOT  9      // 1 + 3 + 5
#define FEAT  144    // C_DIM * MTOT

// workspace layout (float offsets)
#define VMP_OFF  0                      // [B,N,144] message-passed features
#define U_OFF    (VMP_OFF + 147456)     // [B,N,144] mixed nl features
#define SPH_OFF  (U_OFF + 147456)       // [B,N,9]   sph summed over j
#define VSUM_OFF (SPH_OFF + 9216)       // [B,N,144] pre-normalization output
#define NF_OFF   (VSUM_OFF + 147456)    // 3 norm factors
// total floats = NF_OFF + 3 = 451587  (~1.72 MB)

// ---------------- compile-time Clebsch-Gordan table -----------------------------
constexpr double cfact(int n){ double r=1.0; for(int i=2;i<=n;++i) r*=(double)i; return r; }
constexpr double csqrt(double x){
  if (x <= 0.0) return 0.0;
  double g = x > 1.0 ? x : 1.0;
  for (int i = 0; i < 40; ++i) g = 0.5*(g + x/g);
  return g;
}
constexpr double cg_c(int j1,int m1,int j2,int m2,int j3,int m3){
  double pre = csqrt((double)(2*j3+1)*cfact(j3+j1-j2)*cfact(j3-j1+j2)
                     *cfact(j1+j2-j3)/cfact(j1+j2+j3+1));
  pre *= csqrt(cfact(j3+m3)*cfact(j3-m3)*cfact(j1-m1)
               *cfact(j1+m1)*cfact(j2-m2)*cfact(j2+m2));
  int vmin = 0;
  if (j2-j3-m1 > vmin) vmin = j2-j3-m1;
  if (j1-j3+m2 > vmin) vmin = j1-j3+m2;
  int vmax = j1+j2-j3;
  if (j1-m1 < vmax) vmax = j1-m1;
  if (j2+m2 < vmax) vmax = j2+m2;
  double s = 0.0;
  for (int v = vmin; v <= vmax; ++v){
    double den = cfact(v)*cfact(j1+j2-j3-v)*cfact(j1-m1-v)
               * cfact(j2+m2-v)*cfact(j3-j2+m1+v)*cfact(j3-j1-m2+v);
    s += ((v & 1) ? -1.0 : 1.0)/den;
  }
  return pre*s;
}
constexpr int cgIdx(int l1,int l2,int l,int m1,int m2,int k){
  return ((((l1*3+l2)*3+l)*5+m1)*5+m2)*5+k;
}
struct CGTab { float v[3375]; };
constexpr CGTab makeCG(){
  CGTab t{};
  for (int l1=0;l1<3;++l1) for (int l2=0;l2<3;++l2) for (int l=0;l<3;++l){
    int dl = (l1>l2) ? (l1-l2) : (l2-l1);
    if (l < dl || l > l1+l2) continue;
    for (int m1=0;m1<=2*l1;++m1) for (int m2=0;m2<=2*l2;++m2){
      int k = (m1-l1) + (m2-l2) + l;
      if (k >= 0 && k <= 2*l)
        t.v[cgIdx(l1,l2,l,m1,m2,k)] = (float)cg_c(l1,m1-l1,l2,m2-l2,l,k-l);
    }
  }
  return t;
}
constexpr CGTab CGT = makeCG();

constexpr int PAIR_L1c[3][6] = {{0,1,2,0,0,0},{0,1,1,1,2,2},{0,1,1,2,2,2}};
constexpr int PAIR_L2c[3][6] = {{0,1,2,0,0,0},{1,0,1,2,1,2},{2,1,2,0,1,2}};
constexpr int LOFFc[3] = {0,1,4};

// ---------------- K1: message pass  v_mp = conn @ vertices  (f32 WMMA) ----------
__global__ void k1_msgpass(const int* __restrict__ conn,
                           const float* __restrict__ v0,
                           const float* __restrict__ v1,
                           const float* __restrict__ v2,
                           float* __restrict__ vmp){
  int bid = blockIdx.x;
  int colTile = bid % 9; int t = bid / 9;
  int iTile = t & 7;     int b = t >> 3;
  int lane = threadIdx.x;
  int hi   = lane >> 4;
  int lo   = lane & 15;
  int i    = iTile*16 + lo;      // A-matrix row (target node)
  int col  = colTile*16 + lo;    // B-matrix column (feature)
  int c = col / 9, mo = col % 9;
  const float* vp; int wdt; int m;
  if (mo == 0)      { vp = v0; wdt = 1; m = 0; }
  else if (mo < 4)  { vp = v1; wdt = 3; m = mo-1; }
  else              { vp = v2; wdt = 5; m = mo-4; }
  const int* connRow = conn + (b*N_DIM + i)*N_DIM;
  v8f acc = {};
  for (int j0 = 0; j0 < N_DIM; j0 += 4) {
    v2f a, bf;
#pragma unroll
    for (int p = 0; p < 2; ++p) {
      int j = j0 + 2*hi + p;
      a[p]  = (float)connRow[j];
      bf[p] = vp[((b*N_DIM + j)*C_DIM + c)*wdt + m];
    }
    acc = __builtin_amdgcn_wmma_f32_16x16x4_f32(false, a, false, bf,
                                                (short)0, acc, false, false);
  }
#pragma unroll
  for (int v = 0; v < 8; ++v) {
    int nrow = v + 8*hi;
    vmp[(b*N_DIM + iTile*16 + nrow)*FEAT + col] = acc[v];
  }
}

// ---------------- templated CG-square + mix accumulators ------------------------
// A-fragment z[node, t] generated on the fly; all CG structure folds to literals.
template<int L, int MM>
__device__ __forceinline__ v8f nl_accum(const float* __restrict__ xs,
                                        const float* __restrict__ w,
                                        int node, int hi, int cOut){
  v8f acc = {};
  constexpr int NP = (L==0) ? 3 : 6;
#pragma unroll
  for (int pair = 0; pair < NP; ++pair) {
    const int l1 = PAIR_L1c[L][pair];
    const int l2 = PAIR_L2c[L][pair];
    const float* wp = w + pair*256*C_DIM;
    for (int cd0 = 0; cd0 < 256; cd0 += 4) {   // K-loop over one (l1,l2) chunk
      v2f a, bf;
#pragma unroll
      for (int p = 0; p < 2; ++p) {
        int cd = cd0 + 2*hi + p;
        int c = cd >> 4, d = cd & 15;
        float z = 0.0f;
#pragma unroll
        for (int m1 = 0; m1 < 5; ++m1) {
          if (m1 <= 2*l1) {
            int m2 = MM - L + l1 + l2 - m1;    // m-selection rule
            if (m2 >= 0 && m2 <= 2*l2) {
              z += CGT.v[cgIdx(l1,l2,L,m1,m2,MM)]
                 * xs[node*FEAT + c*MTOT + LOFFc[l1] + m1]
                 * xs[node*FEAT + d*MTOT + LOFFc[l2] + m2];
            }
          }
        }
        a[p]  = z;
        bf[p] = wp[cd*C_DIM + cOut];
      }
      acc = __builtin_amdgcn_wmma_f32_16x16x4_f32(false, a, false, bf,
                                                  (short)0, acc, false, false);
    }
  }
  return acc;
}

template<int L, int MM>
__device__ __forceinline__ v8f rel_accum(const float* __restrict__ xs,
                                         const float* __restrict__ ss,
                                         const float* __restrict__ w,
                                         int node, int hi, int cOut){
  v8f acc = {};
  constexpr int NP = (L==0) ? 3 : 6;
#pragma unroll
  for (int pair = 0; pair < NP; ++pair) {
    const int l1 = PAIR_L1c[L][pair];
    const int l2 = PAIR_L2c[L][pair];
    const float* wp = w + pair*16*C_DIM;
#pragma unroll
    for (int c0 = 0; c0 < 16; c0 += 4) {       // chunk of c*d = 16 (sph: 1 channel)
      v2f a, bf;
#pragma unroll
      for (int p = 0; p < 2; ++p) {
        int c = c0 + 2*hi + p;
        float z = 0.0f;
#pragma unroll
        for (int m1 = 0; m1 < 5; ++m1) {
          if (m1 <= 2*l1) {
            int m2 = MM - L + l1 + l2 - m1;
            if (m2 >= 0 && m2 <= 2*l2) {
              z += CGT.v[cgIdx(l1,l2,L,m1,m2,MM)]
                 * xs[node*FEAT + c*MTOT + LOFFc[l1] + m1]
                 * ss[node*MTOT + LOFFc[l2] + m2];
            }
          }
        }
        a[p] = z;
        bf[p] = wp[c*C_DIM + cOut];
      }
      acc = __builtin_amdgcn_wmma_f32_16x16x4_f32(false, a, false, bf,
                                                  (short)0, acc, false, false);
    }
  }
  return acc;
}

// ---------------- K2: fused CG self-square + channel mix (nl) -------------------
// block = 9 waves; wave w handles output (l, mm); 16 nodes per block.
__global__ void k2_nl(const float* __restrict__ vmp,
                      const float* __restrict__ wnl0,
                      const float* __restrict__ wnl1,
                      const float* __restrict__ wnl2,
                      float* __restrict__ u){
  __shared__ float x[16*FEAT];
  int b = blockIdx.x >> 3;
  int nodeBase = (blockIdx.x & 7) * 16;
  const float* src = vmp + (b*N_DIM + nodeBase)*FEAT;
  for (int i = threadIdx.x; i < 16*FEAT; i += blockDim.x) x[i] = src[i];
  __syncthreads();

  int wave = threadIdx.x >> 5;
  int lane = threadIdx.x & 31;
  int node = lane & 15;
  int hi   = lane >> 4;
  int cOut = lane & 15;
  v8f acc;
  int moff;
  switch (wave) {                                  // uniform per wave: EXEC stays all-1s
    case 0:  acc = nl_accum<0,0>(x, wnl0, node, hi, cOut); moff = 0; break;
    case 1:  acc = nl_accum<1,0>(x, wnl1, node, hi, cOut); moff = 1; break;
    case 2:  acc = nl_accum<1,1>(x, wnl1, node, hi, cOut); moff = 2; break;
    case 3:  acc = nl_accum<1,2>(x, wnl1, node, hi, cOut); moff = 3; break;
    case 4:  acc = nl_accum<2,0>(x, wnl2, node, hi, cOut); moff = 4; break;
    case 5:  acc = nl_accum<2,1>(x, wnl2, node, hi, cOut); moff = 5; break;
    case 6:  acc = nl_accum<2,2>(x, wnl2, node, hi, cOut); moff = 6; break;
    case 7:  acc = nl_accum<2,3>(x, wnl2, node, hi, cOut); moff = 7; break;
    default: acc = nl_accum<2,4>(x, wnl2, node, hi, cOut); moff = 8; break;
  }
#pragma unroll
  for (int v = 0; v < 8; ++v) {
    int nrow = v + 8*hi;
    u[(b*N_DIM + nodeBase + nrow)*FEAT + cOut*MTOT + moff] = acc[v];
  }
}

// ---------------- K3: sphsum[b,i,mo] = sum_j sph[b,i,j,mo] ----------------------
__global__ void k3_sphsum(const float* __restrict__ s0,
                          const float* __restrict__ s1,
                          const float* __restrict__ s2,
                          float* __restrict__ sphsum){
  int idx = blockIdx.x*blockDim.x + threadIdx.x;
  if (idx >= B_DIM*N_DIM*MTOT) return;
  int mo = idx % MTOT; int bi = idx / MTOT;
  const float* sp; int wdt; int m;
  if (mo == 0)     { sp = s0; wdt = 1; m = 0; }
  else if (mo < 4) { sp = s1; wdt = 3; m = mo-1; }
  else             { sp = s2; wdt = 5; m = mo-4; }
  const float* base = sp + (size_t)bi*N_DIM*wdt + m;
  float acc = 0.0f;
  for (int j = 0; j < N_DIM; ++j) acc += base[j*wdt];
  sphsum[idx] = acc;
}

// ---------------- K4: fused CG(u, sphsum) + mix_rel (j-sum collapsed) -----------
__global__ void k4_rel(const float* __restrict__ u,
                       const float* __restrict__ sphsum,
                       const float* __restrict__ wr0,
                       const float* __restrict__ wr1,
                       const float* __restrict__ wr2,
                       float* __restrict__ vsum){
  __shared__ float x[16*FEAT];
  __shared__ float s[16*MTOT];
  int b = blockIdx.x >> 3;
  int nodeBase = (blockIdx.x & 7) * 16;
  const float* srcU = u + (b*N_DIM + nodeBase)*FEAT;
  const float* srcS = sphsum + (b*N_DIM + nodeBase)*MTOT;
  for (int i = threadIdx.x; i < 16*FEAT; i += blockDim.x) x[i] = srcU[i];
  for (int i = threadIdx.x; i < 16*MTOT; i += blockDim.x) s[i] = srcS[i];
  __syncthreads();

  int wave = threadIdx.x >> 5;
  int lane = threadIdx.x & 31;
  int node = lane & 15;
  int hi   = lane >> 4;
  int cOut = lane & 15;
  v8f acc;
  int moff;
  switch (wave) {
    case 0:  acc = rel_accum<0,0>(x, s, wr0, node, hi, cOut); moff = 0; break;
    case 1:  acc = rel_accum<1,0>(x, s, wr1, node, hi, cOut); moff = 1; break;
    case 2:  acc = rel_accum<1,1>(x, s, wr1, node, hi, cOut); moff = 2; break;
    case 3:  acc = rel_accum<1,2>(x, s, wr1, node, hi, cOut); moff = 3; break;
    case 4:  acc = rel_accum<2,0>(x, s, wr2, node, hi, cOut); moff = 4; break;
    case 5:  acc = rel_accum<2,1>(x, s, wr2, node, hi, cOut); moff = 5; break;
    case 6:  acc = rel_accum<2,2>(x, s, wr2, node, hi, cOut); moff = 6; break;
    case 7:  acc = rel_accum<2,3>(x, s, wr2, node, hi, cOut); moff = 7; break;
    default: acc = rel_accum<2,4>(x, s, wr2, node, hi, cOut); moff = 8; break;
  }
#pragma unroll
  for (int v = 0; v < 8; ++v) {
    int nrow = v + 8*hi;
    vsum[(b*N_DIM + nodeBase + nrow)*FEAT + cOut*MTOT + moff] = acc[v];
  }
}

// ---------------- K5a: deterministic per-l sum of squares -----------------------
__global__ void k_nf(const float* __restrict__ vsum, float* __restrict__ nf){
  int l = blockIdx.x;
  __shared__ float red[256];
  int width = 2*l + 1;
  int off = (l==0) ? 0 : ((l==1) ? 1 : 4);
  float acc = 0.0f;
  for (int i = threadIdx.x; i < B_DIM*N_DIM*C_DIM; i += 256) {
    const float* p = vsum + i*MTOT + off;
    for (int m = 0; m < width; ++m) { float v = p[m]; acc += v*v; }
  }
  red[threadIdx.x] = acc;
  __syncthreads();
  for (int st = 128; st > 0; st >>= 1) {
    if (threadIdx.x < st) red[threadIdx.x] += red[threadIdx.x + st];
    __syncthreads();
  }
  if (threadIdx.x == 0) nf[l] = red[0];
}

// ---------------- K5b: normalize + write concatenated output --------------------
__global__ void k5_out(const float* __restrict__ vsum, const float* __restrict__ nf,
                       float* __restrict__ out){
  int idx = blockIdx.x*blockDim.x + threadIdx.x;
  if (idx >= B_DIM*N_DIM*C_DIM*MTOT) return;
  int mo = idx % MTOT;
  int l = (mo == 0) ? 0 : ((mo < 4) ? 1 : 2);
  float scale = sqrtf((float)C_DIM / nf[l]);
  out[idx] = vsum[idx] * scale;   // layout matches [B,N,1,C,9] m-concat exactly
}

extern "C" void kernel_launch(void* const* d_in, const int* in_sizes, int n_in,
                              void* d_out, int out_size, void* d_ws, size_t ws_size,
                              hipStream_t stream) {
  (void)in_sizes; (void)n_in; (void)out_size; (void)ws_size;
  const float* v0   = (const float*)d_in[0];
  const float* s0   = (const float*)d_in[1];
  const float* v1   = (const float*)d_in[2];
  const float* s1   = (const float*)d_in[3];
  const float* v2   = (const float*)d_in[4];
  const float* s2   = (const float*)d_in[5];
  const int*   conn = (const int*)  d_in[6];
  const float* wnl0 = (const float*)d_in[7];
  const float* wr0  = (const float*)d_in[8];
  const float* wnl1 = (const float*)d_in[9];
  const float* wr1  = (const float*)d_in[10];
  const float* wnl2 = (const float*)d_in[11];
  const float* wr2  = (const float*)d_in[12];

  float* ws   = (float*)d_ws;
  float* vmp  = ws + VMP_OFF;
  float* u    = ws + U_OFF;
  float* sphs = ws + SPH_OFF;
  float* vsum = ws + VSUM_OFF;
  float* nf   = ws + NF_OFF;
  float* out  = (float*)d_out;

  k1_msgpass<<<B_DIM*8*9, 32, 0, stream>>>(conn, v0, v1, v2, vmp);
  k2_nl     <<<B_DIM*8, 288, 0, stream>>>(vmp, wnl0, wnl1, wnl2, u);
  k3_sphsum <<<(B_DIM*N_DIM*MTOT + 255)/256, 256, 0, stream>>>(s0, s1, s2, sphs);
  k4_rel    <<<B_DIM*8, 288, 0, stream>>>(u, sphs, wr0, wr1, wr2, vsum);
  k_nf      <<<3, 256, 0, stream>>>(vsum, nf);
  k5_out    <<<(B_DIM*N_DIM*C_DIM*MTOT + 255)/256, 256, 0, stream>>>(vsum, nf, out);
}